// ConvBNReLU_37572373905411
// MI455X (gfx1250) — compile-verified
//
#include <hip/hip_runtime.h>
#include <hip/hip_bf16.h>

typedef __attribute__((ext_vector_type(2))) float v2f;
typedef __attribute__((ext_vector_type(8))) float v8f;

#define BATCH 32
#define CIN   32
#define COUT  32
#define HDIM  192
#define WDIM  192
#define KTOT  288          // CIN * 9
#define XCOLS 66           // 64 outputs + 2 halo columns
#define XROW  (3 * XCOLS)  // 198 (col,ky) slots per parity class
#define CPAD  18           // ci/2 dimension: 16 padded to 18 (even, bank-friendly)
#define PLANE (HDIM * WDIM)
#define BN_EPS 1e-5f
#define CUT_THRESH 1e-3f

// ---------------------------------------------------------------------------
// Kernel 1: zero the per-(b,c) plane-max slots in workspace (1024 uints)
// ---------------------------------------------------------------------------
__global__ void init_max_kernel(unsigned int* wsmax) {
    int i = blockIdx.x * blockDim.x + threadIdx.x;
    if (i < BATCH * COUT) wsmax[i] = 0u;   // bit pattern 0 == 0.0f
}

// ---------------------------------------------------------------------------
// Kernel 2: implicit-GEMM conv + bias + BN + ReLU via V_WMMA_F32_16X16X4_F32.
// K ordered as k' = tap*32 + ci.  B operands staged as
//   X2[(p*XROW + ky*66 + col)*18 + ci/2]   (p = ci&1)
// so each lane's per-step B pair is one 8B-aligned ds_load_b64 at a
// compile-time immediate offset from a single per-lane base VGPR.
// ---------------------------------------------------------------------------
__global__ void __launch_bounds__(256)
conv_bn_relu_wmma_kernel(const float* __restrict__ x,
                         const float* __restrict__ conv_w,
                         const float* __restrict__ conv_b,
                         const float* __restrict__ gamma,
                         const float* __restrict__ beta,
                         const float* __restrict__ run_mean,
                         const float* __restrict__ run_var,
                         float* __restrict__ out,
                         unsigned int* __restrict__ wsmax) {
    __shared__ float Wlds[COUT * KTOT];       // 36864 B, [cout][tap*32+ci]
    __shared__ float X2[2 * XROW * CPAD];     // 28512 B, see header comment

    const int tid  = threadIdx.x;
    const int blk  = blockIdx.x;
    const int b    = blk / (HDIM * 3);
    const int rem  = blk % (HDIM * 3);
    const int h    = rem / 3;
    const int w0   = (rem % 3) * 64;

    // ---- stage weights permuted: Wlds[m][tap*32+ci] = conv_w[m][ci][tap].
    // thread (m = tid>>3, p = tid&7) copies 36 contiguous source floats;
    // q/9 and q%9 fold to constants under unroll.
    {
        const int m = tid >> 3;
        const int p = tid & 7;
        const float* wsrc = conv_w + m * KTOT + p * 36;
        float* wdst = Wlds + m * KTOT + 4 * p;       // ci = 4*p + q/9
        #pragma unroll
        for (int q = 0; q < 36; ++q)
            wdst[(q % 9) * CIN + (q / 9)] = wsrc[q];
    }

    // ---- stage input halo tile into X2; threads 0..191, one per half-row
    if (tid < 192) {
        const int r    = tid >> 1;        // 0..95 = ci*3 + ky
        const int ci   = r / 3;
        const int ky   = r % 3;
        const int hh   = h - 1 + ky;
        const bool hok = (hh >= 0) && (hh < HDIM);
        const int c0   = (tid & 1) * 33;  // first col of this half-row
        const float* src = x + ((b * CIN + ci) * HDIM + hh) * WDIM + (w0 - 1);
        float* dst = X2 + ((ci & 1) * XROW + ky * XCOLS) * CPAD + (ci >> 1);
        #pragma unroll
        for (int q = 0; q < 33; ++q) {
            int col = c0 + q;
            int ww  = w0 - 1 + col;
            float v = (hok && ww >= 0 && ww < WDIM) ? src[col] : 0.0f;
            dst[col * CPAD] = v;
        }
    }
    __syncthreads();

    const int wave   = tid >> 5;
    const int lane   = tid & 31;
    const int half   = lane >> 4;       // 0: lanes 0-15, 1: lanes 16-31
    const int nn     = lane & 15;
    const int m_tile = wave & 1;
    const int n_tile = wave >> 1;

    const int m   = m_tile * 16 + nn;        // A-matrix row for this lane
    const int col = n_tile * 16 + nn;        // output pixel within 64-strip

    // Per-lane bases; every LDS access below is base + compile-time offset.
    const float* __restrict__ abase = &Wlds[m * KTOT + 2 * half];
    const float* __restrict__ xb    = &X2[(half * XROW + col) * CPAD];

    v8f acc = {};

    #pragma unroll
    for (int tap = 0; tap < 9; ++tap) {
        const int ky = tap / 3, kx = tap % 3;       // compile-time under unroll
        const int tb = (ky * XCOLS + kx) * CPAD;    // tap base, compile-time
        #pragma unroll
        for (int s = 0; s < 8; ++s) {
            // A 16x4: lanes<16 hold K pair (k0,k0+1); lanes>=16 (k0+2,k0+3)
            v2f a = *reinterpret_cast<const v2f*>(&abase[tap * CIN + 4 * s]);
            // B 4x16: contiguous pair {ci0+half, ci0+2+half} -> one b64 load
            v2f bv = *reinterpret_cast<const v2f*>(&xb[tb + 2 * s]);

            acc = __builtin_amdgcn_wmma_f32_16x16x4_f32(
                /*neg_a=*/false, a, /*neg_b=*/false, bv,
                /*c_mod=*/(short)0, acc, /*reuse_a=*/false, /*reuse_b=*/false);
        }
    }

    // ---- epilogue: bias + BN + ReLU, store, per-plane max reduction
    // D layout: VGPR v -> M = v + 8*half; N = nn  (ISA 7.12.2)
    const int c0   = m_tile * 16 + half * 8;
    const int wcol = w0 + n_tile * 16 + nn;

    #pragma unroll
    for (int v = 0; v < 8; ++v) {
        int c = c0 + v;
        float scale = gamma[c] * rsqrtf(run_var[c] + BN_EPS);
        float val = (acc[v] + conv_b[c] - run_mean[c]) * scale + beta[c];
        val = fmaxf(val, 0.0f);
        out[((b * COUT + c) * HDIM + h) * WDIM + wcol] = val;

        // max over the 16 pixels held by this lane-half (xor<16 stays in half)
        float mx = val;
        mx = fmaxf(mx, __shfl_xor(mx, 1, 32));
        mx = fmaxf(mx, __shfl_xor(mx, 2, 32));
        mx = fmaxf(mx, __shfl_xor(mx, 4, 32));
        mx = fmaxf(mx, __shfl_xor(mx, 8, 32));
        if (nn == 0)   // post-ReLU values >= 0: uint order == float order
            atomicMax(&wsmax[b * COUT + c], __float_as_uint(mx));
    }
}

// ---------------------------------------------------------------------------
// Kernel 3: zero any (b,c) plane whose abs-max < CUT_THRESH
// ---------------------------------------------------------------------------
__global__ void cut_kernel(float* __restrict__ out,
                           const unsigned int* __restrict__ wsmax) {
    int plane = blockIdx.x;               // 0 .. B*COUT-1
    float mx = __uint_as_float(wsmax[plane]);
    if (mx >= CUT_THRESH) return;
    float* p = out + (size_t)plane * PLANE;
    for (int i = threadIdx.x; i < PLANE; i += blockDim.x) p[i] = 0.0f;
}

extern "C" void kernel_launch(void* const* d_in, const int* in_sizes, int n_in,
                              void* d_out, int out_size, void* d_ws, size_t ws_size,
                              hipStream_t stream) {
    const float* x        = (const float*)d_in[0];
    const float* conv_w   = (const float*)d_in[1];
    const float* conv_b   = (const float*)d_in[2];
    const float* gamma    = (const float*)d_in[3];
    const float* beta     = (const float*)d_in[4];
    const float* run_mean = (const float*)d_in[5];
    const float* run_var  = (const float*)d_in[6];
    float* out            = (float*)d_out;
    unsigned int* wsmax   = (unsigned int*)d_ws;   // 1024 uints

    init_max_kernel<<<(BATCH * COUT + 255) / 256, 256, 0, stream>>>(wsmax);

    const int nblocks = BATCH * HDIM * (WDIM / 64);   // 18432
    conv_bn_relu_wmma_kernel<<<nblocks, 256, 0, stream>>>(
        x, conv_w, conv_b, gamma, beta, run_mean, run_var, out, wsmax);

    cut_kernel<<<BATCH * COUT, 256, 0, stream>>>(out, wsmax);
}